// BasicGcn_65189013618896
// MI455X (gfx1250) — compile-verified
//
#include <hip/hip_runtime.h>
#include <math.h>

// ---------------------------------------------------------------------------
// BasicGcn on MI455X (gfx1250, wave32).
// 3x { GEMM (WMMA f32 16x16x4) -> init(out = dinv^2*xl + b) -> edge scatter
//      (atomic f32 adds, L2-resident) -> activation }.
// Degree normalization computed once (graph is layer-invariant).
// ---------------------------------------------------------------------------

typedef __attribute__((ext_vector_type(2))) float v2f;
typedef __attribute__((ext_vector_type(8))) float v8f;

#define GCN_N 50000
#define GCN_E 600000
#define GCN_D 128
#define TILE_K 32
#define LDS_PAD 4   // stride 36 floats -> 4-bank step, conflict free for 16-lane groups

// ---------------------------- degree / norm --------------------------------

__global__ __launch_bounds__(256) void deg_init_k(float* __restrict__ deg, int n) {
    int i = blockIdx.x * 256 + threadIdx.x;
    if (i < n) deg[i] = 1.0f;                 // self-loop weight 1
}

__global__ __launch_bounds__(256) void deg_accum_k(float* __restrict__ deg,
                                                   const int* __restrict__ col,
                                                   const float* __restrict__ w, int e) {
    int i = blockIdx.x * 256 + threadIdx.x;
    if (i < e) atomicAdd(&deg[col[i]], w[i]);
}

__global__ __launch_bounds__(256) void dinv_k(const float* __restrict__ deg,
                                              float* __restrict__ dinv, int n) {
    int i = blockIdx.x * 256 + threadIdx.x;
    if (i < n) {
        float d = deg[i];
        dinv[i] = (d > 0.0f) ? rsqrtf(d) : 0.0f;
    }
}

// ------------------------------- GEMM --------------------------------------
// C[N][128] = A[N][128] @ W[128][128], f32 WMMA 16x16x4.
// Block = 256 threads = 8 waves; block handles a 16-row panel, wave `w`
// handles column tile [16w, 16w+16). K-loop in LDS chunks of 32.
__global__ __launch_bounds__(256) void gemm_wmma_k(const float* __restrict__ A,
                                                   const float* __restrict__ W,
                                                   float* __restrict__ C) {
    __shared__ float As[16][TILE_K + LDS_PAD];        // row-major 16 x 32 chunk
    __shared__ float Wt[GCN_D][TILE_K + LDS_PAD];     // n-major: Wt[n][k] = W[k][n]

    const int t    = threadIdx.x;
    const int lane = t & 31;
    const int wave = t >> 5;                          // 0..7 -> column tile
    const int rowBase = blockIdx.x * 16;              // N = 50000 = 3125 * 16

    v8f acc = {};

    for (int kc = 0; kc < GCN_D; kc += TILE_K) {
        __syncthreads();
        // A chunk: 16x32 = 512 floats (2 per thread), contiguous k -> coalesced
        for (int idx = t; idx < 16 * TILE_K; idx += 256) {
            int m = idx / TILE_K, k = idx % TILE_K;
            As[m][k] = A[(rowBase + m) * GCN_D + kc + k];
        }
        // W chunk transposed: 32x128 = 4096 floats (16 per thread),
        // global read coalesced over n, LDS write n-major.
        for (int idx = t; idx < TILE_K * GCN_D; idx += 256) {
            int k = idx / GCN_D, n = idx % GCN_D;
            Wt[n][k] = W[(kc + k) * GCN_D + n];
        }
        __syncthreads();

        // WMMA f32 16x16x4 lane layout:
        //   A: lanes 0-15 supply K=kk+0,kk+1 ; lanes 16-31 supply K=kk+2,kk+3
        //   B: mirrored ; C/D: VGPR v -> M=v (lanes 0-15) / M=v+8 (lanes 16-31)
        const int m   = lane & 15;
        const int n   = wave * 16 + (lane & 15);
        const int klo = (lane >> 4) * 2;
        #pragma unroll
        for (int kk = 0; kk < TILE_K; kk += 4) {
            v2f a, b;
            a.x = As[m][kk + klo];
            a.y = As[m][kk + klo + 1];
            b.x = Wt[n][kk + klo];
            b.y = Wt[n][kk + klo + 1];
            acc = __builtin_amdgcn_wmma_f32_16x16x4_f32(
                /*neg_a=*/false, a, /*neg_b=*/false, b,
                /*c_mod=*/(short)0, acc, /*reuse_a=*/false, /*reuse_b=*/false);
        }
    }

    const int col    = wave * 16 + (lane & 15);
    const int rowOff = (lane >> 4) * 8;
    #pragma unroll
    for (int v = 0; v < 8; ++v) {
        C[(rowBase + v + rowOff) * GCN_D + col] = acc[v];
    }
}

// ----------------------- self-loop init + bias -----------------------------
// out[i][d] = dinv[i]^2 * xl[i][d] + b[d]
__global__ __launch_bounds__(256) void init_out_k(float* __restrict__ out,
                                                  const float* __restrict__ xl,
                                                  const float* __restrict__ dinv,
                                                  const float* __restrict__ b) {
    int idx = blockIdx.x * 256 + threadIdx.x;     // N*D/256 = 25000 blocks exact
    int i = idx >> 7;
    int d = idx & (GCN_D - 1);
    float di = dinv[i];
    out[idx] = di * di * xl[idx] + b[d];
}

// --------------------------- edge scatter ----------------------------------
// One wave (32 lanes) per edge: float4 gather of source row, 4 atomic adds
// per lane into the target row. norm = dinv[r]*w*dinv[c].
__global__ __launch_bounds__(256) void scatter_k(float* __restrict__ out,
                                                 const float* __restrict__ xl,
                                                 const int* __restrict__ row,
                                                 const int* __restrict__ col,
                                                 const float* __restrict__ w,
                                                 const float* __restrict__ dinv,
                                                 int e_total) {
    int wave = threadIdx.x >> 5;
    int lane = threadIdx.x & 31;
    int e = blockIdx.x * 8 + wave;
    if (e >= e_total) return;                     // wave-uniform guard

    int r = row[e];
    int c = col[e];
    float norm = dinv[r] * w[e] * dinv[c];

    const float4* src = (const float4*)(xl + (size_t)r * GCN_D);
    float4 v = src[lane];                         // 16B per lane, coalesced 512B/row
    float* dst = out + (size_t)c * GCN_D + lane * 4;
    atomicAdd(dst + 0, norm * v.x);
    atomicAdd(dst + 1, norm * v.y);
    atomicAdd(dst + 2, norm * v.z);
    atomicAdd(dst + 3, norm * v.w);
}

// ---------------------------- activations ----------------------------------
// mode 0: ELU ; mode 1: exact-erf GELU ; mode 2: PReLU(scalar slope)
__global__ __launch_bounds__(256) void act_k(float* __restrict__ out, int mode,
                                             const float* __restrict__ prelu_w) {
    int idx = blockIdx.x * 256 + threadIdx.x;
    float h = out[idx];
    float r;
    if (mode == 0) {
        r = (h > 0.0f) ? h : (expf(h) - 1.0f);
    } else if (mode == 1) {
        r = 0.5f * h * (1.0f + erff(h * 0.70710678118654752f));
    } else {
        r = (h >= 0.0f) ? h : prelu_w[0] * h;
    }
    out[idx] = r;
}

// ------------------------------ driver -------------------------------------

extern "C" void kernel_launch(void* const* d_in, const int* in_sizes, int n_in,
                              void* d_out, int out_size, void* d_ws, size_t ws_size,
                              hipStream_t stream) {
    (void)in_sizes; (void)n_in; (void)out_size; (void)ws_size;

    const float* x   = (const float*)d_in[0];
    const int*   row = (const int*)d_in[1];            // edge_index[0], E entries
    const int*   col = ((const int*)d_in[1]) + GCN_E;  // edge_index[1]
    const float* w   = (const float*)d_in[2];
    const float* W1  = (const float*)d_in[3];
    const float* b1  = (const float*)d_in[4];
    const float* W2  = (const float*)d_in[5];
    const float* b2  = (const float*)d_in[6];
    const float* W3  = (const float*)d_in[7];
    const float* b3  = (const float*)d_in[8];
    const float* pw  = (const float*)d_in[9];

    float* out  = (float*)d_out;                       // doubles as inter-layer h
    float* xl   = (float*)d_ws;                        // N*D GEMM result
    float* deg  = xl + (size_t)GCN_N * GCN_D;          // N
    float* dinv = deg + GCN_N;                         // N

    const int nThr = 256;
    const int gN   = (GCN_N + nThr - 1) / nThr;
    const int gE   = (GCN_E + nThr - 1) / nThr;
    const int gND  = (GCN_N * GCN_D) / nThr;           // 25000, exact
    const int gGemm = GCN_N / 16;                      // 3125, exact
    const int gScat = (GCN_E + 7) / 8;                 // 8 edges (waves) per block

    // Normalization (layer-invariant): deg = 1 + sum_w per target, dinv = rsqrt
    deg_init_k <<<gN, nThr, 0, stream>>>(deg, GCN_N);
    deg_accum_k<<<gE, nThr, 0, stream>>>(deg, col, w, GCN_E);
    dinv_k     <<<gN, nThr, 0, stream>>>(deg, dinv, GCN_N);

    // Layer 1: GCNConv -> ELU
    gemm_wmma_k<<<gGemm, nThr, 0, stream>>>(x, W1, xl);
    init_out_k <<<gND,   nThr, 0, stream>>>(out, xl, dinv, b1);
    scatter_k  <<<gScat, nThr, 0, stream>>>(out, xl, row, col, w, dinv, GCN_E);
    act_k      <<<gND,   nThr, 0, stream>>>(out, 0, pw);

    // Layer 2: GCNConv -> GELU(exact)
    gemm_wmma_k<<<gGemm, nThr, 0, stream>>>(out, W2, xl);
    init_out_k <<<gND,   nThr, 0, stream>>>(out, xl, dinv, b2);
    scatter_k  <<<gScat, nThr, 0, stream>>>(out, xl, row, col, w, dinv, GCN_E);
    act_k      <<<gND,   nThr, 0, stream>>>(out, 1, pw);

    // Layer 3: GCNConv -> PReLU
    gemm_wmma_k<<<gGemm, nThr, 0, stream>>>(out, W3, xl);
    init_out_k <<<gND,   nThr, 0, stream>>>(out, xl, dinv, b3);
    scatter_k  <<<gScat, nThr, 0, stream>>>(out, xl, row, col, w, dinv, GCN_E);
    act_k      <<<gND,   nThr, 0, stream>>>(out, 2, pw);
}